// CollectMerge_13048110645917
// MI455X (gfx1250) — compile-verified
//
#include <hip/hip_runtime.h>

#define B_   4
#define P_   9
#define CO_  80
#define H_   128
#define W_   128
#define HW_  (H_ * W_)

// ---------------- CDNA5 async global->LDS wrappers ----------------
#if defined(__has_builtin)
#  if __has_builtin(__builtin_amdgcn_global_load_async_to_lds_b32)
#    define HAVE_ASYNC_B32 1
#  endif
#  if __has_builtin(__builtin_amdgcn_s_wait_asynccnt)
#    define HAVE_WAIT_ASYNC 1
#  endif
#endif

__device__ __forceinline__ void async_copy_f32(const float* src, float* lds_dst) {
#if HAVE_ASYNC_B32
    auto g = (__attribute__((address_space(1))) int*)(src);
    auto l = (__attribute__((address_space(3))) int*)(lds_dst);
    __builtin_amdgcn_global_load_async_to_lds_b32(g, l, /*offset=*/0, /*cpol=*/0);
#else
    *lds_dst = *src;   // synchronous fallback (still correct)
#endif
}

__device__ __forceinline__ void wait_async_all() {
#if HAVE_ASYNC_B32
#  if HAVE_WAIT_ASYNC
    __builtin_amdgcn_s_wait_asynccnt(0);
#  else
    asm volatile("s_wait_asynccnt 0" ::: "memory");
#  endif
#endif
}

// ---------------- Pass 1: transpose (B,P,Co,HW) -> (B,P,HW,Co) ----------------
#define TP_PIX 64   // pixels per tile
#define TP_S   69   // padded LDS row stride in floats (bank-conflict break)

__global__ void CollectMerge_transpose_kernel(const float* __restrict__ x,
                                              float* __restrict__ xt) {
    __shared__ float lds[CO_ * TP_S];

    const int chunks = HW_ / TP_PIX;           // 256
    const int bp     = blockIdx.x / chunks;    // fused (b,p) in [0,36)
    const int pix0   = (blockIdx.x % chunks) * TP_PIX;

    const float* src = x + (size_t)bp * CO_ * HW_ + pix0;

    // Stage 80 x 64 tile into LDS via async global->LDS (coalesced 256B rows).
    for (int i = threadIdx.x; i < CO_ * TP_PIX; i += blockDim.x) {
        const int c = i >> 6;          // / TP_PIX
        const int t = i & (TP_PIX - 1);
        async_copy_f32(src + (size_t)c * HW_ + t, &lds[c * TP_S + t]);
    }
    wait_async_all();
    __syncthreads();

    // Write channels-last: contiguous region of TP_PIX*CO_ floats, float4 stores.
    float* dst = xt + ((size_t)bp * HW_ + pix0) * CO_;
    for (int j = threadIdx.x; j < (CO_ * TP_PIX) / 4; j += blockDim.x) {
        const int f   = j * 4;
        const int pix = f / CO_;
        const int c   = f % CO_;       // CO_ % 4 == 0 -> 4 consecutive channels, same pixel
        float4 v;
        v.x = lds[(c + 0) * TP_S + pix];
        v.y = lds[(c + 1) * TP_S + pix];
        v.z = lds[(c + 2) * TP_S + pix];
        v.w = lds[(c + 3) * TP_S + pix];
        ((float4*)dst)[j] = v;
    }
}

// ---------------- Pass 2: bilinear gather + merge ----------------
// One wave per output pixel. Lane L (L < 20) owns channels 4L..4L+3.
// CHLAST=true : feat is (B,P,HW,Co)   -> one coalesced float4 per lane per tap.
// CHLAST=false: feat is (B,P,Co,HW)   -> strided fallback (ws too small).
template <bool CHLAST>
__global__ void CollectMerge_gather_kernel(const float* __restrict__ feat,
                                           const float* __restrict__ loc,
                                           const float* __restrict__ bias,
                                           float* __restrict__ out) {
    const int lane = threadIdx.x & 31;
    const int wid  = (blockIdx.x * blockDim.x + threadIdx.x) >> 5;
    const int nw   = (gridDim.x * blockDim.x) >> 5;
    const bool act = lane < (CO_ / 4);     // 20 active lanes
    const int  c0  = lane * 4;

    float4 bias4 = make_float4(0.f, 0.f, 0.f, 0.f);
    if (act) bias4 = ((const float4*)bias)[lane];

    for (int gp = wid; gp < B_ * HW_; gp += nw) {
        const int b    = gp >> 14;         // / HW_
        const int opix = gp & (HW_ - 1);

        float4 acc = bias4;

        for (int p = 0; p < P_; ++p) {
            const size_t lb = ((size_t)(b * P_ + p)) * 2 * HW_ + opix;
            const float  y  = loc[lb];
            const float  xx = loc[lb + HW_];

            const float y0f = floorf(y), x0f = floorf(xx);
            const float dy = y - y0f, dx = xx - x0f;
            const int   y0 = (int)y0f, x0 = (int)x0f;
            const int   y1 = y0 + 1,   x1 = x0 + 1;

            const float wy0 = 1.f - dy, wy1 = dy;
            const float wx0 = 1.f - dx, wx1 = dx;

            auto tap = [&](int yi, int xi, float w) {
                const bool valid = (yi >= 0) & (yi < H_) & (xi >= 0) & (xi < W_);
                const int  yc = min(max(yi, 0), H_ - 1);
                const int  xc = min(max(xi, 0), W_ - 1);
                const int  idx = yc * W_ + xc;
                const float wv = valid ? w : 0.f;
                if (act) {
                    if (CHLAST) {
                        const float4 v = ((const float4*)(feat +
                            ((size_t)(b * P_ + p) * HW_ + idx) * CO_))[lane];
                        acc.x += wv * v.x;  acc.y += wv * v.y;
                        acc.z += wv * v.z;  acc.w += wv * v.w;
                    } else {
                        const float* fp = feat +
                            ((size_t)(b * P_ + p) * CO_ + c0) * HW_ + idx;
                        acc.x += wv * fp[0 * HW_];
                        acc.y += wv * fp[1 * HW_];
                        acc.z += wv * fp[2 * HW_];
                        acc.w += wv * fp[3 * HW_];
                    }
                }
            };
            tap(y0, x0, wy0 * wx0);
            tap(y0, x1, wy0 * wx1);
            tap(y1, x0, wy1 * wx0);
            tap(y1, x1, wy1 * wx1);
        }

        if (act) {
            const size_t ob = ((size_t)b * CO_ + c0) * HW_ + opix;
            out[ob + 0 * HW_] = acc.x;
            out[ob + 1 * HW_] = acc.y;
            out[ob + 2 * HW_] = acc.z;
            out[ob + 3 * HW_] = acc.w;
        }
    }
}

extern "C" void kernel_launch(void* const* d_in, const int* in_sizes, int n_in,
                              void* d_out, int out_size, void* d_ws, size_t ws_size,
                              hipStream_t stream) {
    const float* x    = (const float*)d_in[0];   // (B, P*CO, H, W)
    const float* loc  = (const float*)d_in[1];   // (B, 2P, H, W)
    const float* bias = (const float*)d_in[2];   // (CO,)
    float*       out  = (float*)d_out;           // (B, CO, H, W)

    const size_t xt_bytes = (size_t)B_ * P_ * HW_ * CO_ * sizeof(float);

    if (ws_size >= xt_bytes) {
        float* xt = (float*)d_ws;
        CollectMerge_transpose_kernel<<<B_ * P_ * (HW_ / TP_PIX), 256, 0, stream>>>(x, xt);
        CollectMerge_gather_kernel<true><<<4096, 256, 0, stream>>>(xt, loc, bias, out);
    } else {
        CollectMerge_gather_kernel<false><<<4096, 256, 0, stream>>>(x, loc, bias, out);
    }
}